// PeriodicTorsion_8529805049979
// MI455X (gfx1250) — compile-verified
//
#include <hip/hip_runtime.h>
#include <hip/hip_bf16.h>

// Periodic torsion energy: E = sum_t fc[t] * (1 + cos(per[t]*phi[t] - phase[t]))
// phi = signed dihedral angle from 4 gathered coordinates.
//
// MI455X reasoning: 4M torsions stream ~28B each once (indices + params) ->
// ~112MB @ 23.3TB/s ~= 5us floor. coords (2.4MB) fit in the 192MB L2, so the
// 16M random gathers are L2 hits. Memory-stream bound => use NT temporal hints
// on the read-once streams to keep coords resident, prefetch the stream,
// wave32 butterfly reduce + one fp32 global atomic per block.

typedef int v4i __attribute__((ext_vector_type(4)));

__global__ void PeriodicTorsion_zero_kernel(float* out) {
    if (threadIdx.x == 0 && blockIdx.x == 0) out[0] = 0.0f;
}

__device__ __forceinline__ float3 gather3(const float* __restrict__ c, int i) {
    const float* p = c + (size_t)i * 3u;          // -> global_load_b96 (temporal/RT)
    return make_float3(p[0], p[1], p[2]);
}

__global__ __launch_bounds__(256, 8)
void PeriodicTorsion_8529805049979_kernel(const float* __restrict__ coords,
                                          const int*   __restrict__ torsions, // T x 4
                                          const float* __restrict__ fc,
                                          const float* __restrict__ per,
                                          const float* __restrict__ phase,
                                          float*       __restrict__ out,
                                          int T)
{
    const int stride = gridDim.x * blockDim.x;
    float acc = 0.0f;

    for (int t = blockIdx.x * blockDim.x + threadIdx.x; t < T; t += stride) {
        // Prefetch next grid-stride iteration's streaming data (global_prefetch_b8).
        {
            int tn = t + stride;
            if (tn < T) {
                __builtin_prefetch(torsions + 4 * (size_t)tn, 0, 0);
                __builtin_prefetch(fc + tn, 0, 0);
            }
        }

        // Read-once streams: non-temporal so they don't evict the hot coords
        // working set from L2. int4 indices -> single global_load_b128 th:NT.
        v4i   idx = __builtin_nontemporal_load((const v4i*)torsions + t);
        float k   = __builtin_nontemporal_load(fc + t);
        float n   = __builtin_nontemporal_load(per + t);
        float ph  = __builtin_nontemporal_load(phase + t);

        // Gathers: 2.4MB table, keep in cache (regular temporal loads).
        float3 p0 = gather3(coords, idx.x);
        float3 p1 = gather3(coords, idx.y);
        float3 p2 = gather3(coords, idx.z);
        float3 p3 = gather3(coords, idx.w);

        float b1x = p1.x - p0.x, b1y = p1.y - p0.y, b1z = p1.z - p0.z;
        float b2x = p2.x - p1.x, b2y = p2.y - p1.y, b2z = p2.z - p1.z;
        float b3x = p3.x - p2.x, b3y = p3.y - p2.y, b3z = p3.z - p2.z;

        // n1 = b1 x b2, n2 = b2 x b3
        float n1x = b1y * b2z - b1z * b2y;
        float n1y = b1z * b2x - b1x * b2z;
        float n1z = b1x * b2y - b1y * b2x;
        float n2x = b2y * b3z - b2z * b3y;
        float n2y = b2z * b3x - b2x * b3z;
        float n2z = b2x * b3y - b2y * b3x;

        float d12 = n1x * n2x + n1y * n2y + n1z * n2z;
        float q1  = n1x * n1x + n1y * n1y + n1z * n1z;
        float q2  = n2x * n2x + n2y * n2y + n2z * n2z;

        float cosv = d12 * __frsqrt_rn(q1 * q2);       // v_rsq_f32
        cosv = fminf(fmaxf(cosv, -0.999999999f), 0.999999999f);

        float s   = n1x * b3x + n1y * b3y + n1z * b3z;
        float sgn = (float)(s > 0.0f) - (float)(s < 0.0f);
        float phi = acosf(cosv) * sgn;

        acc += k * (1.0f + __cosf(n * phi - ph));      // v_cos_f32 path
    }

    // wave32 butterfly reduction (ds_swizzle/permlane on gfx1250)
    #pragma unroll
    for (int off = 16; off > 0; off >>= 1)
        acc += __shfl_xor(acc, off, 32);

    __shared__ float wsum[8];                          // 256 threads = 8 wave32s
    const int lane = threadIdx.x & 31;
    const int wave = threadIdx.x >> 5;
    if (lane == 0) wsum[wave] = acc;
    __syncthreads();

    if (wave == 0) {
        float v = (lane < 8) ? wsum[lane] : 0.0f;
        #pragma unroll
        for (int off = 4; off > 0; off >>= 1)
            v += __shfl_xor(v, off, 32);
        if (lane == 0) atomicAdd(out, v);              // global_atomic_add_f32
    }
}

extern "C" void kernel_launch(void* const* d_in, const int* in_sizes, int n_in,
                              void* d_out, int out_size, void* d_ws, size_t ws_size,
                              hipStream_t stream) {
    (void)n_in; (void)out_size; (void)d_ws; (void)ws_size;

    const float* coords   = (const float*)d_in[0];
    const int*   torsions = (const int*)  d_in[1];
    const float* fc       = (const float*)d_in[2];
    const float* per      = (const float*)d_in[3];
    const float* phase    = (const float*)d_in[4];
    float*       out      = (float*)d_out;

    const int T = in_sizes[2];                         // fc has one entry per torsion

    PeriodicTorsion_zero_kernel<<<1, 1, 0, stream>>>(out);

    int blocks = (T + 255) / 256;
    if (blocks > 16384) blocks = 16384;                // grid-stride loop covers rest
    if (blocks < 1) blocks = 1;

    PeriodicTorsion_8529805049979_kernel<<<blocks, 256, 0, stream>>>(
        coords, torsions, fc, per, phase, out, T);
}